// ReverseCostExtractor_7739531067975
// MI455X (gfx1250) — compile-verified
//
#include <hip/hip_runtime.h>

// CDNA5 / gfx1250, wave32. One wave32 workgroup per output pixel (16384 blocks).
//
// Stage 1: 100 bilinear gathers (4 corners each) from per-map 64x64 cost maps
//          into a 10x10 LDS window V (zero-padded to 12x16 for WMMA K/N pad).
//          Gather loop is divergence-free (clamped indices + mult. masks) so
//          all 16 global_load_b32 per lane can be in flight together.
// Stage 2: Out(9x9) = Ly(9x10) * V(10x10) * Rx^T(10x9) done as two
//          V_WMMA_F32_16X16X4_F32 chains (3 k-steps each), full f32 precision.
//          Banded weights built branchlessly (v_cndmask, no exec juggling).

typedef __attribute__((ext_vector_type(2))) float v2f;
typedef __attribute__((ext_vector_type(8))) float v8f;

// Banded interpolation matrix element:
//   band[m][k] = (m<9) * ( (k==m) ? (1-w) : (k==m+1) ? w : 0 )
__device__ __forceinline__ float band(int k, int mm, float w, bool live) {
    float r = (k == mm)     ? (1.0f - w) : 0.0f;
    r       = (k == mm + 1) ? w          : r;
    return live ? r : 0.0f;
}

__global__ __launch_bounds__(32) void rce_kernel(
    const float* __restrict__ cost_maps,  // (B*64*64, 1, 64, 64) f32
    const float* __restrict__ coords0,    // (B, 2, 64, 64) f32, ch0=x ch1=y
    const float* __restrict__ coords1,    // (B, 2, 64, 64) f32
    float* __restrict__ out)              // (B, 81, 64, 64) f32
{
    const int p    = blockIdx.x;      // pixel id: b*4096 + i1*64 + j1
    const int b    = p >> 12;
    const int rem  = p & 4095;        // i1*64 + j1
    const int lane = threadIdx.x;     // 0..31 (wave32)

    // ---- uniform per-pixel scalars -------------------------------------
    const int cbase = b * 2 * 4096 + rem;
    const float cx = coords0[cbase];          // stage-2 centroid x
    const float cy = coords0[cbase + 4096];   // stage-2 centroid y
    const float x1 = coords1[cbase];          // stage-1 sample x
    const float y1 = coords1[cbase + 4096];   // stage-1 sample y

    const float fx0 = floorf(cx), fy0 = floorf(cy);
    const float wx0 = cx - fx0,   wy0 = cy - fy0;
    const int   ix0 = (int)fx0,   iy0 = (int)fy0;

    const float fx1 = floorf(x1), fy1 = floorf(y1);
    const float wx1 = x1 - fx1,   wy1 = y1 - fy1;
    const int   jx  = (int)fx1,   jy  = (int)fy1;

    // stage-1 corner validity folded into the 4 weights (zero-padding sampler)
    const float vx0 = ((unsigned)jx       < 64u) ? 1.0f : 0.0f;
    const float vx1 = ((unsigned)(jx + 1) < 64u) ? 1.0f : 0.0f;
    const float vy0 = ((unsigned)jy       < 64u) ? 1.0f : 0.0f;
    const float vy1 = ((unsigned)(jy + 1) < 64u) ? 1.0f : 0.0f;
    const float w00 = (1.0f - wy1) * (1.0f - wx1) * vy0 * vx0;
    const float w01 = (1.0f - wy1) * wx1          * vy0 * vx1;
    const float w10 = wy1          * (1.0f - wx1) * vy1 * vx0;
    const float w11 = wy1          * wx1          * vy1 * vx1;

    const int cjx0 = min(max(jx, 0), 63),     cjx1 = min(max(jx + 1, 0), 63);
    const int cjy0 = min(max(jy, 0), 63),     cjy1 = min(max(jy + 1, 0), 63);
    const int o00 = cjy0 * 64 + cjx0, o01 = cjy0 * 64 + cjx1;
    const int o10 = cjy1 * 64 + cjx0, o11 = cjy1 * 64 + cjx1;

    // ---- LDS tiles -----------------------------------------------------
    __shared__ float sV[12][16];   // V window, rows 0..9 / cols 0..9 live, rest 0
    __shared__ float sT[16][17];   // T = Ly*V (stride 17: no bank conflict)

    // Zero padding region (same-wave LDS ops are in-order; no barrier needed
    // before the partial overwrite below).
    for (int i = lane; i < 12 * 16; i += 32) ((float*)sV)[i] = 0.0f;

    // ---- Stage 1: 100 window gathers, no divergence --------------------
    // Lanes with t >= 100 redo point 99 (same value, same address: benign).
    const size_t mapbase = (size_t)b * 4096u;   // map id: b*4096 + my*64 + mx
#pragma unroll
    for (int it = 0; it < 4; ++it) {
        const int t  = min(lane + it * 32, 99);
        const int ty = t / 10, tx = t - ty * 10;
        const int my = iy0 - 4 + ty;          // stage-2 zero padding on [0,64)
        const int mx = ix0 - 4 + tx;
        const float vmask =
            ((unsigned)my < 64u && (unsigned)mx < 64u) ? 1.0f : 0.0f;
        const int myc = min(max(my, 0), 63);
        const int mxc = min(max(mx, 0), 63);
        const float* mp =
            cost_maps + ((mapbase + (size_t)(myc * 64 + mxc)) << 12);
        const float V =
            vmask * (w00 * mp[o00] + w01 * mp[o01] +
                     w10 * mp[o10] + w11 * mp[o11]);
        sV[ty][tx] = V;
    }
    __syncthreads();

    // ---- Stage 2a: T = Ly * V via WMMA f32 16x16x4 (K padded 10->12) ---
    // A 16x4 layout: lane m=L%16, VGPR v -> K = v + 2*(L/16)
    // B 4x16 layout: VGPR v, K = v + 2*(L/16), N = L%16
    const int  m    = lane & 15;
    const int  hl   = lane >> 4;
    const bool live = (m < 9);

    v8f accT = {0.f, 0.f, 0.f, 0.f, 0.f, 0.f, 0.f, 0.f};
#pragma unroll
    for (int s = 0; s < 3; ++s) {
        v2f a, bm;
#pragma unroll
        for (int v = 0; v < 2; ++v) {
            const int k = 4 * s + v + 2 * hl;     // 0..11
            a[v]  = band(k, m, wy0, live);         // Ly[m][k]
            bm[v] = sV[k][m];                      // V[k][x=m]
        }
        accT = __builtin_amdgcn_wmma_f32_16x16x4_f32(
            false, a, false, bm, (short)0, accT, false, false);
    }

    // D layout: VGPR r, lane L -> row r + 8*(L/16), col L%16
#pragma unroll
    for (int r = 0; r < 8; ++r) sT[r + 8 * hl][m] = accT[r];
    __syncthreads();

    // ---- Stage 2b: Out = T * Rx^T via WMMA f32 16x16x4 -----------------
    v8f accO = {0.f, 0.f, 0.f, 0.f, 0.f, 0.f, 0.f, 0.f};
#pragma unroll
    for (int s = 0; s < 3; ++s) {
        v2f a2, b2;
#pragma unroll
        for (int v = 0; v < 2; ++v) {
            const int k = 4 * s + v + 2 * hl;     // 0..11
            a2[v] = sT[m][k];                      // T[row m][col k]
            b2[v] = band(k, m, wx0, live);         // Rx^T[k][n=m]
        }
        accO = __builtin_amdgcn_wmma_f32_16x16x4_f32(
            false, a2, false, b2, (short)0, accO, false, false);
    }

    // ---- Scatter 81 outputs: out[b, kk=a*9+bb, i1, j1] -----------------
    float* ob = out + (size_t)b * 81u * 4096u + (size_t)rem;
#pragma unroll
    for (int r = 0; r < 8; ++r) {
        const int bb = r + 8 * hl;    // y-offset index
        const int aa = m;             // x-offset index
        if (bb < 9 && aa < 9) {
            ob[(size_t)(aa * 9 + bb) * 4096u] = accO[r];
        }
    }
}

extern "C" void kernel_launch(void* const* d_in, const int* in_sizes, int n_in,
                              void* d_out, int out_size, void* d_ws, size_t ws_size,
                              hipStream_t stream) {
    const float* cost_maps = (const float*)d_in[0];
    const float* coords0   = (const float*)d_in[1];
    const float* coords1   = (const float*)d_in[2];
    float*       out       = (float*)d_out;

    const int BHW = in_sizes[1] / 2;   // coords0 has B*2*H*W elements -> 16384 pixels
    rce_kernel<<<BHW, 32, 0, stream>>>(cost_maps, coords0, coords1, out);
}